// stfgcn_32401233281404
// MI455X (gfx1250) — compile-verified
//
#include <hip/hip_runtime.h>
#include <math.h>

typedef __attribute__((ext_vector_type(2))) float v2f;
typedef __attribute__((ext_vector_type(8))) float v8f;

#define NB 4
#define TT 12
#define NN 883
#define FF 152
#define F2 304

// ---------------------------------------------------------------------------
// WMMA inner loop: A from global (contiguous float2 per lane), B from LDS
// (weights pre-transposed to fragment order, contiguous float2 per lane).
// aPtr = &A[rowc*lda + kb], wPtr = &Wt[col*Ktot + ktap + kb], kb = 2*(lane>=16)
// ---------------------------------------------------------------------------
__device__ __forceinline__ v8f wmma_ld(const float* __restrict__ aPtr, float ascale,
                                       const float* __restrict__ wPtr, int K, v8f acc)
{
#pragma unroll 4
  for (int k = 0; k < K; k += 4) {
    v2f a = *(const v2f*)(aPtr + k);   // global_load_b64
    v2f b = *(const v2f*)(wPtr + k);   // ds_load_b64
    a.x *= ascale;
    a.y *= ascale;
    acc = __builtin_amdgcn_wmma_f32_16x16x4_f32(false, a, false, b, (short)0, acc,
                                                false, false);
  }
  return acc;
}

__device__ __forceinline__ v8f zero8() {
  v8f z; for (int i = 0; i < 8; ++i) z[i] = 0.0f; return z;
}

// ---------------------------------------------------------------------------
// ex = concat([x[...,:3]@in_proj + b, tod_emb[idx], dow_emb[idx], adp_emb], -1)
// ---------------------------------------------------------------------------
__global__ void k_ex(const float* __restrict__ x, const float* __restrict__ inw,
                     const float* __restrict__ inb, const float* __restrict__ tod,
                     const float* __restrict__ dow, const float* __restrict__ adp,
                     float* __restrict__ ex)
{
  int bn = blockIdx.x;             // (b*T + t)*N + n
  int f  = threadIdx.x;
  if (f >= FF) return;
  int n  = bn % NN;
  int bt = bn / NN;
  int t  = bt % TT;
  const float* xr = x + (size_t)bn * 3;
  float val;
  if (f < 24) {
    val = inb[f];
    for (int k = 0; k < 3; ++k) val += xr[k] * inw[k * 24 + f];
  } else if (f < 48) {
    int idx = (int)(xr[1] * 288.0f);
    val = tod[idx * 24 + (f - 24)];
  } else if (f < 72) {
    int idx = (int)xr[2];
    val = dow[idx * 24 + (f - 48)];
  } else {
    val = adp[((size_t)t * NN + n) * 80 + (f - 72)];
  }
  ex[(size_t)bn * FF + f] = val;
}

// ---------------------------------------------------------------------------
// corr graph, diagonal-only closed form (rank-1 Q/K; beta cancels in softmax)
// ---------------------------------------------------------------------------
__global__ void k_corr(const float* __restrict__ xsrc, int xstride,
                       const float* __restrict__ wq, const float* __restrict__ bq,
                       const float* __restrict__ wk, const float* __restrict__ bk,
                       float* __restrict__ sc)
{
  __shared__ float xm[NN];
  __shared__ float rmin[256], rmax[256];
  int bt  = blockIdx.y;
  int tid = threadIdx.x;
  const float* xp = xsrc + (size_t)bt * NN * xstride;
  float lmin = 1e30f, lmax = -1e30f;
  for (int m = tid; m < NN; m += 256) {
    float v = xp[(size_t)m * xstride];
    xm[m] = v;
    lmin = fminf(lmin, v);
    lmax = fmaxf(lmax, v);
  }
  rmin[tid] = lmin; rmax[tid] = lmax;
  __syncthreads();
  for (int off = 128; off; off >>= 1) {
    if (tid < off) {
      rmin[tid] = fminf(rmin[tid], rmin[tid + off]);
      rmax[tid] = fmaxf(rmax[tid], rmax[tid + off]);
    }
    __syncthreads();
  }
  float mnX = rmin[0], mxX = rmax[0];
  float a = 0.0f, cc = 0.0f;
  for (int d = 0; d < 32; ++d) { a += wq[d] * wk[d]; cc += bq[d] * wk[d]; }
  const float inv = 0.17677669529663687f;  // 1/sqrt(32)
  int n = blockIdx.x * 256 + tid;
  if (n < NN) {
    float xn    = xm[n];
    float alpha = (a * xn + cc) * inv;
    float mx    = (alpha >= 0.0f) ? alpha * mxX : alpha * mnX;
    float ssum  = 0.0f;
    for (int m = 0; m < NN; ++m) ssum += expf(alpha * xm[m] - mx);
    sc[(size_t)bt * NN + n] = expf(alpha * xn - mx) / ssum;
  }
}

// g[b*N+n] = cb + sum_t cw[t]*sc[(b*T+t)*N+n]
__global__ void k_gsum(const float* __restrict__ sc, const float* __restrict__ cw,
                       const float* __restrict__ cb, float* __restrict__ g)
{
  int i = blockIdx.x * blockDim.x + threadIdx.x;
  if (i >= NB * NN) return;
  int b = i / NN, n = i % NN;
  float acc = cb[0];
  for (int t = 0; t < TT; ++t) acc += cw[t] * sc[((size_t)b * TT + t) * NN + n];
  g[i] = acc;
}

// dg = 0.5*(g1 + relu(sigmoid(g1)*g2))
__global__ void k_diagmix(const float* __restrict__ g1, const float* __restrict__ g2,
                          float* __restrict__ dg)
{
  int i = blockIdx.x * blockDim.x + threadIdx.x;
  if (i >= NB * NN) return;
  float d = g1[i], e = g2[i];
  float sig = 1.0f / (1.0f + expf(-d));
  dg[i] = 0.5f * (d + fmaxf(0.0f, sig * e));
}

// ---------------------------------------------------------------------------
// conv1 (F->2F, width 2) + bias + exact GELU.
// block = 4 waves; wave w does row tile blockIdx.x*4+w; weights staged in LDS.
// grid: (14, 19, B*5)
// ---------------------------------------------------------------------------
__global__ void k_conv1(const float* __restrict__ ex, const float* __restrict__ w,
                        const float* __restrict__ bias, float* __restrict__ out1, int s)
{
  __shared__ float Wt[16 * 2 * FF];                 // [col][tap][k], 19.4 KB
  int to = blockIdx.y, bw = blockIdx.z;
  int b = bw / 5, wp = bw % 5;
  // stage weights: Wt[col*304 + tap*152 + kk] = c1_w[((to*16+col)*152+kk)*2 + tap]
  for (int idx = threadIdx.x; idx < 16 * 2 * FF; idx += 128) {
    int col = idx / (2 * FF);
    int rem = idx - col * (2 * FF);
    int tap = rem / FF;
    int kk  = rem - tap * FF;
    Wt[idx] = w[(size_t)((to * 16 + col) * FF + kk) * 2 + tap];
  }
  __syncthreads();
  int wave = threadIdx.x >> 5;
  int lane = threadIdx.x & 31;
  int tn   = blockIdx.x * 4 + wave;
  int r    = lane & 15;
  int kb   = (lane >> 4) * 2;
  int row  = tn * 16 + r;
  int rowc = row < NN ? row : NN - 1;
  int col  = to * 16 + r;                           // O=304, always in range
  v8f acc = zero8();
  for (int tap = 0; tap < 2; ++tap) {
    const float* A = ex + (((size_t)(b * TT + s + wp + tap)) * NN + rowc) * FF + kb;
    const float* W = &Wt[r * (2 * FF) + tap * FF + kb];
    acc = wmma_ld(A, 1.0f, W, FF, acc);
  }
  float* C = out1 + ((size_t)(b * 5 + wp)) * NN * F2;
  int rh = (lane >> 4) * 8;
  for (int j = 0; j < 8; ++j) {
    int rr = tn * 16 + rh + j;
    if (rr < NN) {
      float v = acc[j] + bias[col];
      v = 0.5f * v * (1.0f + erff(v * 0.70710678118654752f));   // exact GELU
      C[(size_t)rr * F2 + col] = v;
    }
  }
}

// conv2 (2F->F, width 2) + bias. grid: (14, 10, B*4) -> yd (pre-LN)
__global__ void k_conv2(const float* __restrict__ o1, const float* __restrict__ w,
                        const float* __restrict__ bias, float* __restrict__ yd)
{
  __shared__ float Wt[16 * 2 * F2];                 // 38.9 KB
  int to = blockIdx.y, bw = blockIdx.z;
  int b = bw / 4, wp = bw % 4;
  for (int idx = threadIdx.x; idx < 16 * 2 * F2; idx += 128) {
    int col = idx / (2 * F2);
    int rem = idx - col * (2 * F2);
    int tap = rem / F2;
    int kk  = rem - tap * F2;
    int o   = to * 16 + col;
    int oc  = o < FF ? o : FF - 1;
    Wt[idx] = w[(size_t)(oc * F2 + kk) * 2 + tap];
  }
  __syncthreads();
  int wave = threadIdx.x >> 5;
  int lane = threadIdx.x & 31;
  int tn   = blockIdx.x * 4 + wave;
  int r    = lane & 15;
  int kb   = (lane >> 4) * 2;
  int row  = tn * 16 + r;
  int rowc = row < NN ? row : NN - 1;
  int col  = to * 16 + r;
  int colc = col < FF ? col : FF - 1;
  v8f acc = zero8();
  for (int tap = 0; tap < 2; ++tap) {
    const float* A = o1 + (((size_t)(b * 5 + wp + tap)) * NN + rowc) * F2 + kb;
    const float* W = &Wt[r * (2 * F2) + tap * F2 + kb];
    acc = wmma_ld(A, 1.0f, W, F2, acc);
  }
  float* C = yd + ((size_t)(b * 4 + wp)) * NN * FF;
  int rh = (lane >> 4) * 8;
  for (int j = 0; j < 8; ++j) {
    int rr = tn * 16 + rh + j;
    if (rr < NN && col < FF) C[(size_t)rr * FF + col] = acc[j] + bias[colc];
  }
}

// c3 (F->F, width 6) + bias + ReLU. grid: (14, 10, B) -> ys (pre-LN)
__global__ void k_conv3t(const float* __restrict__ ex, const float* __restrict__ w,
                         const float* __restrict__ bias, float* __restrict__ ys, int s)
{
  __shared__ float Wt[16 * 6 * FF];                 // 58.4 KB
  int to = blockIdx.y, b = blockIdx.z;
  for (int idx = threadIdx.x; idx < 16 * 6 * FF; idx += 128) {
    int col = idx / (6 * FF);
    int rem = idx - col * (6 * FF);
    int tap = rem / FF;
    int kk  = rem - tap * FF;
    int o   = to * 16 + col;
    int oc  = o < FF ? o : FF - 1;
    Wt[idx] = w[(size_t)(oc * FF + kk) * 6 + tap];
  }
  __syncthreads();
  int wave = threadIdx.x >> 5;
  int lane = threadIdx.x & 31;
  int tn   = blockIdx.x * 4 + wave;
  int r    = lane & 15;
  int kb   = (lane >> 4) * 2;
  int row  = tn * 16 + r;
  int rowc = row < NN ? row : NN - 1;
  int col  = to * 16 + r;
  int colc = col < FF ? col : FF - 1;
  v8f acc = zero8();
  for (int tap = 0; tap < 6; ++tap) {
    const float* A = ex + (((size_t)(b * TT + s + tap)) * NN + rowc) * FF + kb;
    const float* W = &Wt[r * (6 * FF) + tap * FF + kb];
    acc = wmma_ld(A, 1.0f, W, FF, acc);
  }
  float* C = ys + (size_t)b * NN * FF;
  int rh = (lane >> 4) * 8;
  for (int j = 0; j < 8; ++j) {
    int rr = tn * 16 + rh + j;
    if (rr < NN && col < FF)
      C[(size_t)rr * FF + col] = fmaxf(acc[j] + bias[colc], 0.0f);
  }
}

// LayerNorm over last dim (F=152), in place; one wave per row.
__global__ void k_ln(float* __restrict__ data, const float* __restrict__ g,
                     const float* __restrict__ b)
{
  int row  = blockIdx.x;
  int lane = threadIdx.x;
  float* p = data + (size_t)row * FF;
  float s = 0.0f, sq = 0.0f;
  for (int f = lane; f < FF; f += 32) { float v = p[f]; s += v; sq += v * v; }
  for (int o = 16; o; o >>= 1) { s += __shfl_xor(s, o, 32); sq += __shfl_xor(sq, o, 32); }
  float mean = s * (1.0f / FF);
  float var  = sq * (1.0f / FF) - mean * mean;
  float inv  = rsqrtf(var + 1e-5f);
  for (int f = lane; f < FF; f += 32) p[f] = (p[f] - mean) * inv * g[f] + b[f];
}

// h[b][3p+j][n][f] = ex[...] + fc_b[j] + fc_w[j][0]*ys + sum_c fc_w[j][1+c]*yd[c]
__global__ void k_combine(const float* __restrict__ ex, const float* __restrict__ ys,
                          const float* __restrict__ yd, const float* __restrict__ fcw,
                          const float* __restrict__ fcb, float* __restrict__ h, int p)
{
  int idx = blockIdx.x * blockDim.x + threadIdx.x;
  if (idx >= NB * 3 * NN * FF) return;
  int f = idx % FF;  int rest = idx / FF;
  int n = rest % NN; rest /= NN;
  int j = rest % 3;  int b = rest / 3;
  int t = p * 3 + j;
  float acc = fcb[j] + fcw[j * 5] * ys[((size_t)b * NN + n) * FF + f];
  for (int c = 0; c < 4; ++c)
    acc += fcw[j * 5 + 1 + c] * yd[(((size_t)b * 4 + c) * NN + n) * FF + f];
  size_t off = (((size_t)b * TT + t) * NN + n) * FF + f;
  h[off] = ex[off] + acc;
}

// xg = relu((h*diag) @ theta3) + h ; diag fused into A fragment.
// grid: (14, 10, B*T)
__global__ void k_theta(const float* __restrict__ h, const float* __restrict__ th,
                        const float* __restrict__ dg, float* __restrict__ xg)
{
  __shared__ float Wt[16 * FF];                     // 9.7 KB
  int to = blockIdx.y, bt = blockIdx.z;
  int b = bt / TT;
  for (int idx = threadIdx.x; idx < 16 * FF; idx += 128) {
    int col = idx / FF;
    int kk  = idx - col * FF;
    int o   = to * 16 + col;
    int oc  = o < FF ? o : FF - 1;
    Wt[idx] = th[(size_t)kk * FF + oc];             // theta3[m][o] -> Wt[col][m]
  }
  __syncthreads();
  int wave = threadIdx.x >> 5;
  int lane = threadIdx.x & 31;
  int tn   = blockIdx.x * 4 + wave;
  int r    = lane & 15;
  int kb   = (lane >> 4) * 2;
  int row  = tn * 16 + r;
  int rowc = row < NN ? row : NN - 1;
  int col  = to * 16 + r;
  const float* A = h + ((size_t)bt * NN + rowc) * FF + kb;
  const float* W = &Wt[r * FF + kb];
  float dsc = dg[(size_t)b * NN + rowc];
  v8f acc = wmma_ld(A, dsc, W, FF, zero8());
  int rh = (lane >> 4) * 8;
  for (int j = 0; j < 8; ++j) {
    int rr = tn * 16 + rh + j;
    if (rr < NN && col < FF) {
      size_t off = ((size_t)bt * NN + rr) * FF + col;
      xg[off] = fmaxf(acc[j], 0.0f) + h[off];
    }
  }
}

// out[b][t'][n] = sum_{t,f} xg[b][t][n][f] * out_w[(t*F+f)*12 + t'] + out_b[t']
__global__ void k_out(const float* __restrict__ xg, const float* __restrict__ ow,
                      const float* __restrict__ ob, float* __restrict__ out)
{
  int bn = blockIdx.x;
  int b = bn / NN, n = bn % NN;
  int lane = threadIdx.x;
  float acc[TT];
  for (int j = 0; j < TT; ++j) acc[j] = 0.0f;
  for (int t = 0; t < TT; ++t) {
    const float* xr = xg + (((size_t)b * TT + t) * NN + n) * FF;
    for (int f = lane; f < FF; f += 32) {
      float xv = xr[f];
      const float* wr = ow + ((size_t)t * FF + f) * TT;
      for (int j = 0; j < TT; ++j) acc[j] += xv * wr[j];
    }
  }
  for (int o = 16; o; o >>= 1)
    for (int j = 0; j < TT; ++j) acc[j] += __shfl_xor(acc[j], o, 32);
  if (lane == 0)
    for (int j = 0; j < TT; ++j)
      out[((size_t)b * TT + j) * NN + n] = acc[j] + ob[j];
}

// ---------------------------------------------------------------------------
struct Params {
  const float *x, *xb, *inw, *inb, *tod, *dow, *adp;
  const float *wq1, *bq1, *wk1, *bk1, *cw1, *cb1;
  const float *wq2, *bq2, *wk2, *bk2, *cw2, *cb2;
  const float *theta3, *ow, *ob;
  const float *c1w[4], *c1b[4], *c2w[4], *c2b[4], *l1g[4], *l1b[4];
  const float *c3w[4], *c3b[4], *l2g[4], *l2b[4], *fcw[4], *fcb[4];
};

extern "C" void kernel_launch(void* const* d_in, const int* in_sizes, int n_in,
                              void* d_out, int out_size, void* d_ws, size_t ws_size,
                              hipStream_t stream)
{
  (void)n_in; (void)out_size; (void)ws_size;
  auto F = [&](int i) { return (const float*)d_in[i]; };
  Params P;
  if (in_sizes[0] == NB * TT * NN * 3) {
    // dict insertion order: x, x_backday, params{...}
    P.x = F(0); P.xb = F(1);
    P.inw = F(2); P.inb = F(3); P.tod = F(4); P.dow = F(5); P.adp = F(6);
    P.wq1 = F(7);  P.bq1 = F(8);  P.wk1 = F(9);  P.bk1 = F(10); P.cw1 = F(11); P.cb1 = F(12);
    P.wq2 = F(13); P.bq2 = F(14); P.wk2 = F(15); P.bk2 = F(16); P.cw2 = F(17); P.cb2 = F(18);
    P.theta3 = F(19);
    for (int p = 0; p < 4; ++p) {
      int o = 20 + p * 12;
      P.c1w[p] = F(o + 0);  P.c1b[p] = F(o + 1);
      P.c2w[p] = F(o + 2);  P.c2b[p] = F(o + 3);
      P.l1g[p] = F(o + 4);  P.l1b[p] = F(o + 5);
      P.c3w[p] = F(o + 6);  P.c3b[p] = F(o + 7);
      P.l2g[p] = F(o + 8);  P.l2b[p] = F(o + 9);
      P.fcw[p] = F(o + 10); P.fcb[p] = F(o + 11);
    }
    P.ow = F(68); P.ob = F(69);
  } else {
    // pytree (sorted-key) order: params first, keys alphabetical
    P.adp = F(0);
    P.bk1 = F(1); P.bq1 = F(2); P.cb1 = F(3); P.cw1 = F(4); P.wk1 = F(5); P.wq1 = F(6);
    P.bk2 = F(7); P.bq2 = F(8); P.cb2 = F(9); P.cw2 = F(10); P.wk2 = F(11); P.wq2 = F(12);
    P.dow = F(13); P.inb = F(14); P.inw = F(15); P.ob = F(16); P.ow = F(17);
    for (int p = 0; p < 4; ++p) {
      int o = 18 + p * 12;
      P.c1b[p] = F(o + 0);  P.c1w[p] = F(o + 1);
      P.c2b[p] = F(o + 2);  P.c2w[p] = F(o + 3);
      P.c3b[p] = F(o + 4);  P.c3w[p] = F(o + 5);
      P.fcb[p] = F(o + 6);  P.fcw[p] = F(o + 7);
      P.l1b[p] = F(o + 8);  P.l1g[p] = F(o + 9);
      P.l2b[p] = F(o + 10); P.l2g[p] = F(o + 11);
    }
    P.theta3 = F(66); P.tod = F(67); P.x = F(68); P.xb = F(69);
  }

  // workspace carve-up (floats)
  const size_t EXSZ = (size_t)NB * TT * NN * FF;   // 6,442,368
  const size_t O1SZ = (size_t)NB * 5 * NN * F2;    // 5,368,640
  const size_t YDSZ = (size_t)NB * 4 * NN * FF;    // 2,147,456
  const size_t YSSZ = (size_t)NB * NN * FF;        //   536,864
  float* ws = (float*)d_ws;
  float* ex = ws;
  float* h  = ex + EXSZ;
  float* o1 = h  + EXSZ;
  float* yd = o1 + O1SZ;
  float* ys = yd + YDSZ;
  float* g1 = ys + YSSZ;
  float* g2 = g1 + 4096;
  float* dg = g2 + 4096;
  float* sc = dg + 4096;                           // NB*TT*NN = 42,384

  float* out = (float*)d_out;

  // 1. build ex
  k_ex<<<NB * TT * NN, 160, 0, stream>>>(P.x, P.inw, P.inb, P.tod, P.dow, P.adp, ex);

  // 2. corr graphs -> diagonal
  dim3 cg((NN + 255) / 256, NB * TT);
  k_corr<<<cg, 256, 0, stream>>>(P.x, 3, P.wq1, P.bq1, P.wk1, P.bk1, sc);
  k_gsum<<<(NB * NN + 255) / 256, 256, 0, stream>>>(sc, P.cw1, P.cb1, g1);
  k_corr<<<cg, 256, 0, stream>>>(P.xb, 1, P.wq2, P.bq2, P.wk2, P.bk2, sc);
  k_gsum<<<(NB * NN + 255) / 256, 256, 0, stream>>>(sc, P.cw2, P.cb2, g2);
  k_diagmix<<<(NB * NN + 255) / 256, 256, 0, stream>>>(g1, g2, dg);

  // 3. TCN patches (sequential; scratch reused). 14 blocks x 4 waves = 56 row tiles
  for (int p = 0; p < 4; ++p) {
    int s = 2 * p;
    k_conv1<<<dim3(14, 19, NB * 5), 128, 0, stream>>>(ex, P.c1w[p], P.c1b[p], o1, s);
    k_conv2<<<dim3(14, 10, NB * 4), 128, 0, stream>>>(o1, P.c2w[p], P.c2b[p], yd);
    k_ln<<<NB * 4 * NN, 32, 0, stream>>>(yd, P.l1g[p], P.l1b[p]);
    k_conv3t<<<dim3(14, 10, NB), 128, 0, stream>>>(ex, P.c3w[p], P.c3b[p], ys, s);
    k_ln<<<NB * NN, 32, 0, stream>>>(ys, P.l2g[p], P.l2b[p]);
    k_combine<<<(NB * 3 * NN * FF + 255) / 256, 256, 0, stream>>>(
        ex, ys, yd, P.fcw[p], P.fcb[p], h, p);
  }

  // 4. xg = relu((h*diag)@theta3) + h   (xg overlays ex, no longer needed)
  k_theta<<<dim3(14, 10, NB * TT), 128, 0, stream>>>(h, P.theta3, dg, ex);

  // 5. output projection
  k_out<<<NB * NN, 32, 0, stream>>>(ex, P.ow, P.ob, out);
}